// BaseRoIHeads_12704513261786
// MI455X (gfx1250) — compile-verified
//
#include <hip/hip_runtime.h>
#include <hip/hip_bf16.h>
#include <math.h>

#define B_IMGS 16
#define NPROP  2048
#define NTILES 128            // NPROP / 16
#define DETS   300
#define IMG_W_ 1333.0f
#define IMG_H_ 800.0f
#define SCORE_T 0.05f
#define NMS_T   0.5f
#define XFORM_CLIP 4.135166556742356f   // log(1000/16)

typedef float v2f __attribute__((ext_vector_type(2)));
typedef float v8f __attribute__((ext_vector_type(8)));
typedef unsigned v4u __attribute__((ext_vector_type(4)));
typedef int v4i __attribute__((ext_vector_type(4)));
typedef int v8i __attribute__((ext_vector_type(8)));

// ---------------------------------------------------------------- zero output
__global__ void k_zero(float* __restrict__ out, int n) {
    int i = blockIdx.x * blockDim.x + threadIdx.x;
    if (i < n) out[i] = 0.0f;
}

// ------------------------------------------------- decode + softmax + clip
__global__ void k_decode(const float* __restrict__ logits,
                         const float* __restrict__ reg,
                         const float* __restrict__ prop,
                         float* __restrict__ boxes_raw,
                         float* __restrict__ scores_raw) {
    int p = blockIdx.x * blockDim.x + threadIdx.x;
    if (p >= B_IMGS * NPROP) return;
    __builtin_prefetch(&reg[p * 8], 0, 0);   // global_prefetch_b8

    float l0 = logits[p * 2 + 0];
    float l1 = logits[p * 2 + 1];
    float sc = 1.0f / (1.0f + __expf(l0 - l1));   // softmax class-1 prob

    float x1 = prop[p * 4 + 0], y1 = prop[p * 4 + 1];
    float x2 = prop[p * 4 + 2], y2 = prop[p * 4 + 3];
    float w  = x2 - x1, h = y2 - y1;
    float cx = x1 + 0.5f * w, cy = y1 + 0.5f * h;

    // class-1 regression columns 4..7 ; weights (10,10,5,5)
    float dx = reg[p * 8 + 4] * 0.1f;
    float dy = reg[p * 8 + 5] * 0.1f;
    float dw = fminf(reg[p * 8 + 6] * 0.2f, XFORM_CLIP);
    float dh = fminf(reg[p * 8 + 7] * 0.2f, XFORM_CLIP);

    float pcx = dx * w + cx, pcy = dy * h + cy;
    float pw  = __expf(dw) * w, ph = __expf(dh) * h;

    float bx1 = fminf(fmaxf(pcx - 0.5f * pw, 0.0f), IMG_W_);
    float by1 = fminf(fmaxf(pcy - 0.5f * ph, 0.0f), IMG_H_);
    float bx2 = fminf(fmaxf(pcx + 0.5f * pw, 0.0f), IMG_W_);
    float by2 = fminf(fmaxf(pcy + 0.5f * ph, 0.0f), IMG_H_);

    boxes_raw[p * 4 + 0] = bx1;
    boxes_raw[p * 4 + 1] = by1;
    boxes_raw[p * 4 + 2] = bx2;
    boxes_raw[p * 4 + 3] = by2;
    scores_raw[p] = sc;
}

// ------------------------------------------- per-image bitonic sort (desc)
__global__ void __launch_bounds__(1024)
k_sort(const float* __restrict__ scores_raw,
       float* __restrict__ sorted_scores,
       const float* __restrict__ boxes_raw,
       float* __restrict__ sorted_boxes) {
    int img = blockIdx.x;
    int tid = threadIdx.x;                 // 0..1023
    __shared__ float s[NPROP];
    __shared__ int   idx[NPROP];

    for (int i = tid; i < NPROP; i += 1024) {
        s[i]  = scores_raw[img * NPROP + i];
        idx[i] = i;
    }
    __syncthreads();

    for (int k = 2; k <= NPROP; k <<= 1) {
        for (int j = k >> 1; j > 0; j >>= 1) {
            for (int i = tid; i < NPROP; i += 1024) {
                int ixj = i ^ j;
                if (ixj > i) {
                    bool upper = (i & k) == 0;       // descending sort
                    float si = s[i], sj = s[ixj];
                    bool sw = upper ? (si < sj) : (si > sj);
                    if (sw) {
                        s[i] = sj; s[ixj] = si;
                        int t = idx[i]; idx[i] = idx[ixj]; idx[ixj] = t;
                    }
                }
            }
            __syncthreads();
        }
    }

    for (int i = tid; i < NPROP; i += 1024) {
        sorted_scores[img * NPROP + i] = s[i];
        int src = idx[i];
        float4 bb = ((const float4*)boxes_raw)[img * NPROP + src];
        ((float4*)sorted_boxes)[img * NPROP + i] = bb;
    }
}

// -------------------- IoU suppression masks: one wave per 16-row stripe.
// TDM async-DMAs the image's 2048 sorted boxes (32KB) into LDS once, then
// the wave sweeps 128 column tiles. Per tile, V_WMMA_F32_16X16X4_F32 forms
// the union's area term  D[M][N] = area_a[M] + area_b[N]  via rank-2 product
// A(16x4) x B(4x16):  A=[area_a,1,0,0], B=[1;area_b;0;0].
__global__ void __launch_bounds__(32)
k_mask(const float* __restrict__ sboxes,
       unsigned short* __restrict__ masks) {
    int img     = blockIdx.y;
    int rowBase = blockIdx.x * 16;
    int lane    = threadIdx.x;
    bool lo     = lane < 16;

    __shared__ float colboxes[NPROP * 4];         // 32 KB image box cache

    // ---- Tensor Data Mover descriptor (D#): 1-D tensor, 8192 x f32 tile
    unsigned long long gaddr =
        (unsigned long long)(const void*)(sboxes + (size_t)img * NPROP * 4);
    unsigned ldsoff = (unsigned)(size_t)(void*)&colboxes[0];

    v4u g0;
    g0[0] = 1u;                                        // count=1, no gather
    g0[1] = ldsoff;                                    // lds_addr
    g0[2] = (unsigned)(gaddr & 0xFFFFFFFFull);         // global_addr[31:0]
    g0[3] = (unsigned)((gaddr >> 32) & 0x01FFFFFFull)  // global_addr[56:32]
            | (2u << 30);                              // type = 2 ("image")

    v8i g1;
    g1[0] = (int)(2u << 16);                 // workgroup_mask=0, data_size=4B
    g1[1] = (int)((unsigned)(NPROP * 4) << 16);  // tensor_dim0[15:0] @63:48
    g1[2] = (int)(1u << 16);                 // tensor_dim0 msbs=0, tensor_dim1=1
    g1[3] = (int)((unsigned)(NPROP * 4) << 16);  // tile_dim0 = 8192 @127:112
    g1[4] = 1;                               // tile_dim1 = 1, tile_dim2 = 0
    g1[5] = NPROP * 4;                       // tensor_dim0_stride[31:0]
    g1[6] = 0;
    g1[7] = 0;
    v4i gz4 = {0, 0, 0, 0};
    v8i gz8 = {0, 0, 0, 0, 0, 0, 0, 0};

    // clang-23 6-arg form: (g0, g1, g2, g3, extra group, cpol)
    __builtin_amdgcn_tensor_load_to_lds(g0, g1, gz4, gz4, gz8, 0); // TENSORcnt++

    // row boxes / A-matrix operand (loop-invariant, overlaps the DMA)
    const float4* bxg = (const float4*)sboxes + (size_t)img * NPROP;
    float4 ba_l = bxg[rowBase + (lane & 15)];
    float area_a = (ba_l.z - ba_l.x) * (ba_l.w - ba_l.y);
    v2f A;
    A.x = lo ? area_a : 0.0f;                // lanes 0-15: K=0,1 ; 16-31: K=2,3
    A.y = lo ? 1.0f   : 0.0f;

    int rowOff = lo ? 0 : 8;
    float4 ba[8];
#pragma unroll
    for (int k = 0; k < 8; ++k) ba[k] = bxg[rowBase + rowOff + k];

    __builtin_amdgcn_s_wait_tensorcnt(0);    // LDS tile ready (single-wave WG)

    const float4* bxl = (const float4*)colboxes;
    for (int ct = 0; ct < NTILES; ++ct) {
        float4 bb = bxl[ct * 16 + (lane & 15)];          // ds_load_b128
        float area_b = (bb.z - bb.x) * (bb.w - bb.y);
        v2f Bm;
        Bm.x = lo ? 1.0f   : 0.0f;
        Bm.y = lo ? area_b : 0.0f;
        v8f C = {0.f,0.f,0.f,0.f,0.f,0.f,0.f,0.f};

        // D[k] : lanes 0-15 -> row M=k ; lanes 16-31 -> M=k+8 ; N = lane&15
        v8f D = __builtin_amdgcn_wmma_f32_16x16x4_f32(
            false, A, false, Bm, (short)0, C, false, false);

#pragma unroll
        for (int k = 0; k < 8; ++k) {
            float ix1 = fmaxf(ba[k].x, bb.x);
            float iy1 = fmaxf(ba[k].y, bb.y);
            float ix2 = fminf(ba[k].z, bb.z);
            float iy2 = fminf(ba[k].w, bb.w);
            float iw  = fmaxf(ix2 - ix1, 0.0f);
            float ih  = fmaxf(iy2 - iy1, 0.0f);
            float inter = iw * ih;
            float uni   = D[k] - inter;      // area_a + area_b - inter >= 0
            // iou > T  <=>  inter > T*uni   (uni>=0; degenerate case -> false)
            bool pred = inter > NMS_T * uni;

            unsigned word = __builtin_amdgcn_ballot_w32(pred);
            if ((lane & 15) == k) {
                unsigned m16 = lo ? (word & 0xFFFFu) : (word >> 16);
                int r = rowBase + rowOff + k;
                masks[((unsigned)img * NPROP + (unsigned)r) * NTILES
                      + (unsigned)ct] = (unsigned short)m16;
            }
        }
    }
}

// ------------------------- greedy NMS scan + top-300 emit: one wave per image
__global__ void __launch_bounds__(32)
k_nms_emit(const float* __restrict__ sboxes,
           const float* __restrict__ sscores,
           const unsigned short* __restrict__ masks,
           float* __restrict__ out) {
    int img  = blockIdx.x;
    int lane = threadIdx.x;
    volatile __shared__ unsigned keepw[NTILES];   // 16 live-bits per tile word

    const float* sc = sscores + (size_t)img * NPROP;
    for (int T = lane; T < NTILES; T += 32) {
        unsigned w = 0;
#pragma unroll
        for (int b = 0; b < 16; ++b)
            if (sc[T * 16 + b] > SCORE_T) w |= (1u << b);
        keepw[T] = w;
    }
    // single wave: LDS ops are in-order within the wave, no barrier needed

    const unsigned short* mrow = masks + (size_t)img * NPROP * NTILES;
    for (int i = 0; i < NPROP; ++i) {
        unsigned wi = keepw[i >> 4];
        if (!((wi >> (i & 15)) & 1u)) continue;   // uniform across wave
        const unsigned short* m = mrow + (size_t)i * NTILES;
        for (int T = lane; T < NTILES; T += 32) {
            int tbase = T * 16;
            if (tbase + 15 <= i) continue;        // no j > i in this tile
            unsigned allowed;
            if (tbase > i) {
                allowed = 0xFFFFu;
            } else {
                int shift = i - tbase + 1;        // 1..15 here
                allowed = (shift >= 16) ? 0u : ((0xFFFFu << shift) & 0xFFFFu);
            }
            unsigned sup = (unsigned)m[T] & allowed;
            if (sup) keepw[T] = keepw[T] & ~sup;
        }
    }

    if (lane == 0) {
        float* ob = out;                                   // [16][300][4]
        float* os = out + B_IMGS * DETS * 4;               // [16][300]
        int*   ol = (int*)(out + B_IMGS * DETS * 5);       // [16][300] int32
        int cnt = 0;
        for (int i = 0; i < NPROP && cnt < DETS; ++i) {
            if ((keepw[i >> 4] >> (i & 15)) & 1u) {
                float4 bb = ((const float4*)sboxes)[(size_t)img * NPROP + i];
                int o = img * DETS + cnt;
                ob[o * 4 + 0] = bb.x;
                ob[o * 4 + 1] = bb.y;
                ob[o * 4 + 2] = bb.z;
                ob[o * 4 + 3] = bb.w;
                os[o] = sc[i];
                ol[o] = 1;
                ++cnt;
            }
        }
    }
}

// ---------------------------------------------------------------------------
extern "C" void kernel_launch(void* const* d_in, const int* in_sizes, int n_in,
                              void* d_out, int out_size, void* d_ws, size_t ws_size,
                              hipStream_t stream) {
    const float* logits = (const float*)d_in[0];   // [B*N, 2]
    const float* reg    = (const float*)d_in[1];   // [B*N, 8]
    const float* prop   = (const float*)d_in[2];   // [B, N, 4]

    float* ws = (float*)d_ws;
    float* boxes_raw  = ws;                         // 131072 f
    float* scores_raw = ws + 131072;                //  32768 f
    float* sboxes     = ws + 163840;                // 131072 f
    float* sscores    = ws + 294912;                //  32768 f
    unsigned short* masks = (unsigned short*)(ws + 327680);  // 16*2048*128 u16

    float* out = (float*)d_out;

    k_zero<<<(out_size + 255) / 256, 256, 0, stream>>>(out, out_size);
    k_decode<<<(B_IMGS * NPROP) / 256, 256, 0, stream>>>(logits, reg, prop,
                                                         boxes_raw, scores_raw);
    k_sort<<<B_IMGS, 1024, 0, stream>>>(scores_raw, sscores, boxes_raw, sboxes);
    dim3 gmask(NTILES, B_IMGS);                    // one wave per 16-row stripe
    k_mask<<<gmask, 32, 0, stream>>>(sboxes, masks);
    k_nms_emit<<<B_IMGS, 32, 0, stream>>>(sboxes, sscores, masks, out);
}